// NeuronTurboModel_40819369181759
// MI455X (gfx1250) — compile-verified
//
#include <hip/hip_runtime.h>

typedef __attribute__((ext_vector_type(2))) float v2f;
typedef __attribute__((ext_vector_type(8))) float v8f;

#define MIN_DIST 1.0f   // RADIUS(0.5) * 2 * scale(1.0)
#define ROW_SPLIT 4     // blocks per tile-row pair (occupancy vs reuse)

// Raw v_sqrt_f32: ~1 ulp, valid for our normal-range inputs (d2 in [0,36]);
// avoids the ~15-instruction IEEE refinement sequence the libcall emits.
__device__ __forceinline__ float fast_sqrt(float v) {
  return __builtin_amdgcn_sqrtf(v);
}

// ---------------------------------------------------------------- reductions
__device__ __forceinline__ float wave_reduce(float v) {
#pragma unroll
  for (int off = 16; off > 0; off >>= 1)
    v += __shfl_xor(v, off, 32);
  return v;
}

// ---------------------------------------------------------------- init
// acc[0] = sum(dist)         (S2)
// acc[1] = sum relu(md - d)  over all pairs (S3)
// acc[2] = sum relu(md - d)  over masked prefix runs (S4, subtracted)
// acc[3] = fixed + 10*noise  (S1 + S5)
__global__ void init_acc(double* acc) {
  if (threadIdx.x < 4) acc[threadIdx.x] = 0.0;
}

// ---------------------------------------------------------------- O(N) terms
__global__ void setup_kernel(const float* __restrict__ x,
                             const float* __restrict__ y,
                             const float* __restrict__ z,
                             double* __restrict__ acc, int n) {
  int i = blockIdx.x * blockDim.x + threadIdx.x;
  float local = 0.f;
  if (i < n) {
    float yi = y[i], zi = z[i];
    // PufferZone on p2d = (y, z), bounds [-1, 1]
    local += fmaxf(yi - 1.f, 0.f) + fmaxf(-1.f - yi, 0.f);
    local += fmaxf(zi - 1.f, 0.f) + fmaxf(-1.f - zi, 0.f);
    // noise: squared second differences, weight 10
    if (i >= 1 && i <= n - 2) {
      float d2y = y[i + 1] - 2.f * yi + y[i - 1];
      float d2z = z[i + 1] - 2.f * zi + z[i - 1];
      local += 10.f * (d2y * d2y + d2z * d2z);
    }
  }
  local = wave_reduce(local);
  if ((threadIdx.x & 31) == 0)
    atomicAdd(&acc[3], (double)local);
}

// ---------------------------------------------------------------- N^2 tiles
// Upper-triangular tiles only (weight 2 off-diagonal, 1 on the tile diagonal).
// Tile-row p is paired with tile-row ntiles-1-p so every pair holds exactly
// ntiles+1 tiles (constant work). Each pair is split across ROW_SPLIT blocks;
// each wave walks a strided strip of tiles, keeping the WMMA A-fragment and
// the 8 row norms resident in VGPRs across the strip (row changes <= 2/wave),
// and issues a single pair of f64 atomics at the end.
__global__ void tile_kernel(const float* __restrict__ x,
                            const float* __restrict__ y,
                            const float* __restrict__ z,
                            double* __restrict__ acc, int n, int ntiles) {
  int p   = blockIdx.x / ROW_SPLIT;          // pair index
  int sub = blockIdx.x % ROW_SPLIT;          // split within pair
  int waveInBlk = threadIdx.x >> 5;
  int lane = threadIdx.x & 31;
  int hi = lane >> 4;                        // 0: K=0,1 (x,y)   1: K=2,3 (z,0)
  int lo = lane & 15;                        // A: row M   B: col N

  int rowA = p;
  int rowB = ntiles - 1 - p;
  int cntA = ntiles - rowA;                            // tiles J=rowA..ntiles-1
  int cntB = (rowB > rowA) ? (ntiles - rowB) : 0;      // tiles J=rowB..ntiles-1
  int total = cntA + cntB;                             // == ntiles+1 (even n)

  const int waves_per_pair = 8 * ROW_SPLIT;
  float sum_d = 0.f, sum_r = 0.f;
  int curRow = -1;
  v2f a = {0.f, 0.f};
  float ni[8];

  for (int t = sub * 8 + waveInBlk; t < total; t += waves_per_pair) {
    int tI, tJ;
    if (t < cntA) { tI = rowA; tJ = rowA + t; }
    else          { tI = rowB; tJ = rowB + (t - cntA); }
    int i0 = tI << 4, j0 = tJ << 4;

    if (tI != curRow) {                      // (re)load A fragment + row norms
      curRow = tI;
      if (hi == 0) { a.x = x[i0 + lo]; a.y = y[i0 + lo]; }
      else         { a.x = z[i0 + lo]; a.y = 0.f; }
#pragma unroll
      for (int v = 0; v < 8; ++v) {
        int i = i0 + v + (hi << 3);          // row m = v + 8*hi
        float xi = x[i], yi = y[i], zi = z[i];
        ni[v] = xi * xi + yi * yi + zi * zi;
      }
    }

    v2f b;
    if (hi == 0) { b.x = x[j0 + lo]; b.y = y[j0 + lo]; }
    else         { b.x = z[j0 + lo]; b.y = 0.f; }

    // prefetch this wave's next B tile (global_prefetch_b8; hint only)
    int jn = j0 + (waves_per_pair << 4);
    if (jn + lo < n) {
      __builtin_prefetch(&x[jn + lo], 0, 1);
      __builtin_prefetch(&y[jn + lo], 0, 1);
      __builtin_prefetch(&z[jn + lo], 0, 1);
    }

    v8f c = {};
    // D = A(16x4) * B(4x16) + 0 : Gram tile G[m,n] = <P_i0+m, P_j0+n>
    c = __builtin_amdgcn_wmma_f32_16x16x4_f32(false, a, false, b,
                                              (short)0, c, false, false);

    float xj = x[j0 + lo], yj = y[j0 + lo], zj = z[j0 + lo];
    float nj = xj * xj + yj * yj + zj * zj;

    float td = 0.f, tr = 0.f;
#pragma unroll
    for (int v = 0; v < 8; ++v) {
      float d2 = fmaxf(ni[v] + nj - 2.f * c[v], 0.f);
      float d = fast_sqrt(d2);               // v_sqrt_f32; sqrt(0)=0
      td += d;
      tr += fmaxf(MIN_DIST - d, 0.f);
    }
    float w = (tI == tJ) ? 1.f : 2.f;        // symmetry weight
    sum_d += w * td;
    sum_r += w * tr;
  }

  sum_d = wave_reduce(sum_d);
  sum_r = wave_reduce(sum_r);
  if (lane == 0) {
    atomicAdd(&acc[0], (double)sum_d);
    atomicAdd(&acc[1], (double)sum_r);
  }
}

// ---------------------------------------------------------------- prefix runs
// Row i: scan j = i, i+1, ... while dist <= min_dist; those entries were
// overwritten with min_dist+1 in the reference, so subtract their relu terms.
__global__ void run_kernel(const float* __restrict__ x,
                           const float* __restrict__ y,
                           const float* __restrict__ z,
                           double* __restrict__ acc, int n) {
  int i = blockIdx.x * blockDim.x + threadIdx.x;
  float s = 0.f;
  if (i < n) {
    float xi = x[i], yi = y[i], zi = z[i];
    for (int j = i; j < n; ++j) {
      float dx = x[j] - xi, dy = y[j] - yi, dz = z[j] - zi;
      float d = fast_sqrt(dx * dx + dy * dy + dz * dz);
      if (d > MIN_DIST) break;
      s += MIN_DIST - d;
    }
  }
  s = wave_reduce(s);
  if ((threadIdx.x & 31) == 0)
    atomicAdd(&acc[2], (double)s);
}

// ---------------------------------------------------------------- combine
__global__ void final_kernel(const double* __restrict__ acc,
                             float* __restrict__ out) {
  if (threadIdx.x == 0 && blockIdx.x == 0)
    out[0] = (float)(acc[3] + acc[0] / 10000.0 + acc[1] - acc[2]);
}

// ---------------------------------------------------------------- launch
extern "C" void kernel_launch(void* const* d_in, const int* in_sizes, int n_in,
                              void* d_out, int out_size, void* d_ws, size_t ws_size,
                              hipStream_t stream) {
  const float* x = (const float*)d_in[0];
  const float* y = (const float*)d_in[1];
  const float* z = (const float*)d_in[2];
  float* out = (float*)d_out;
  int n = in_sizes[0];                       // 8192
  double* acc = (double*)d_ws;               // 4 doubles of scratch

  init_acc<<<1, 32, 0, stream>>>(acc);

  int blk = 256;
  setup_kernel<<<(n + blk - 1) / blk, blk, 0, stream>>>(x, y, z, acc, n);

  int ntiles = n / 16;                       // 512
  int pairs = (ntiles + 1) / 2;              // 256 row pairs, 513 tiles each
  tile_kernel<<<pairs * ROW_SPLIT, 256, 0, stream>>>(x, y, z, acc, n, ntiles);

  run_kernel<<<(n + blk - 1) / blk, blk, 0, stream>>>(x, y, z, acc, n);

  final_kernel<<<1, 32, 0, stream>>>(acc, out);
}